// CoxPartialLikelihoodLoss_72739566125706
// MI455X (gfx1250) — compile-verified
//
#include <hip/hip_runtime.h>
#include <stdint.h>

// ---------------------------------------------------------------------------
// CoxPartialLikelihoodLoss on MI455X (gfx1250, wave32)
//
// Pipeline (all deterministic, no float global atomics):
//   1) key build:    key = ~orderable(time)  -> ascending key == descending time,
//                    stable ties == ascending original index (matches argsort(-t))
//   2) 4-pass LSD radix sort (8-bit digits) of (key, index):
//        hist -> exclusive scan (digit-major, tile-minor) -> stable ballot scatter
//   3) inclusive prefix sum of exp(score) over REVERSED sorted order
//        (== suffix cumsum over sorted order), via WMMA f32 16x16x4 tile scans;
//        input staged with GLOBAL_LOAD_ASYNC_TO_LDS (ASYNCcnt path): a B128
//        block DMA for the index tile + per-lane B32 async *gathers* of
//        scores/events straight into LDS (both arrays are L2-resident).
//   4) masked log-sum reduction -> loss
//
// Workspace need: 4*N*4 (keys+idx ping-pong) + 4MB hist + ~49KB  ≈ 261 MB.
// N = 16,777,216 is exactly divisible by TILE=4096.
// ---------------------------------------------------------------------------

typedef __attribute__((ext_vector_type(2))) float v2f;
typedef __attribute__((ext_vector_type(8))) float v8f;

#define RADIX    256
#define TILE     4096
#define THREADS  256
#define EPT      16
#define EPSF     1e-8f

__device__ __forceinline__ uint32_t time_key(float t) {
  uint32_t b = __float_as_uint(t);
  uint32_t u = (b & 0x80000000u) ? ~b : (b | 0x80000000u); // ascending-float order
  return ~u;                                               // invert -> descending time
}

__device__ __forceinline__ uint32_t lds_off(const void* p) {
  // generic LDS pointer: low 32 bits are the workgroup-relative LDS byte offset
  return (uint32_t)(uintptr_t)p;
}

// ---------------------------- 1) key build ---------------------------------
__global__ __launch_bounds__(THREADS) void k_init_keys(
    const float* __restrict__ times, uint32_t* __restrict__ keys,
    uint32_t* __restrict__ idx, int n) {
  int i = blockIdx.x * THREADS + threadIdx.x;
  if (i < n) {
    keys[i] = time_key(times[i]);
    idx[i]  = (uint32_t)i;
  }
}

// ---------------------------- 2a) histogram --------------------------------
__global__ __launch_bounds__(THREADS) void k_radix_hist(
    const uint32_t* __restrict__ keys, uint32_t* __restrict__ hist,
    int numTiles, int shift) {
  __shared__ uint32_t lh[RADIX];
  int tid = threadIdx.x;
  lh[tid] = 0u;
  __syncthreads();
  int base = blockIdx.x * TILE;
#pragma unroll
  for (int k = 0; k < EPT; ++k) {
    uint32_t d = (keys[base + tid + k * THREADS] >> shift) & 0xFFu;
    atomicAdd(&lh[d], 1u);
  }
  __syncthreads();
  // digit-major, tile-minor layout for a single flat exclusive scan
  hist[tid * numTiles + blockIdx.x] = lh[tid];
}

// ------------------- 2b) generic u32 exclusive scan (3 kernels) ------------
__global__ __launch_bounds__(THREADS) void k_scan_u32_tiles(
    uint32_t* __restrict__ a, uint32_t* __restrict__ tileSums) {
  __shared__ uint32_t tsum[THREADS];
  int tid  = threadIdx.x;
  int base = blockIdx.x * TILE + tid * EPT;
  uint32_t v[EPT];
#pragma unroll
  for (int j = 0; j < EPT; ++j) v[j] = a[base + j];
  uint32_t s = 0;
#pragma unroll
  for (int j = 0; j < EPT; ++j) { uint32_t x = v[j]; v[j] = s; s += x; }
  tsum[tid] = s;
  __syncthreads();
  for (int off = 1; off < THREADS; off <<= 1) {
    uint32_t t = (tid >= off) ? tsum[tid - off] : 0u;
    __syncthreads();
    tsum[tid] += t;
    __syncthreads();
  }
  uint32_t texcl = tsum[tid] - s;
  if (tid == THREADS - 1) tileSums[blockIdx.x] = tsum[THREADS - 1];
#pragma unroll
  for (int j = 0; j < EPT; ++j) a[base + j] = texcl + v[j];
}

__global__ void k_scan_u32_tops(uint32_t* __restrict__ a, int count) {
  if (threadIdx.x == 0) {
    uint32_t run = 0;
    for (int i = 0; i < count; ++i) { uint32_t t = a[i]; a[i] = run; run += t; }
  }
}

__global__ __launch_bounds__(THREADS) void k_scan_u32_add(
    uint32_t* __restrict__ a, const uint32_t* __restrict__ tops) {
  uint32_t add = tops[blockIdx.x];
  int base = blockIdx.x * TILE + threadIdx.x;
#pragma unroll
  for (int k = 0; k < EPT; ++k) a[base + k * THREADS] += add;
}

// --------------------- 2c) stable ballot scatter ---------------------------
__global__ __launch_bounds__(THREADS) void k_radix_scatter(
    const uint32_t* __restrict__ srcK, const uint32_t* __restrict__ srcI,
    uint32_t* __restrict__ dstK, uint32_t* __restrict__ dstI,
    const uint32_t* __restrict__ histScanned, int numTiles, int shift) {
  __shared__ uint32_t wdc[8][RADIX];   // per-wave running digit counts
  __shared__ uint32_t exW[8][RADIX];   // cross-wave exclusive digit prefix
  int tid  = threadIdx.x;
  int wave = tid >> 5;
  int lane = tid & 31;
#pragma unroll
  for (int j = 0; j < 8; ++j) wdc[j][tid & (RADIX - 1)] = 0u;  // 8*256 init
  __syncthreads();

  int base = blockIdx.x * TILE;
  uint32_t myKey[EPT], myPay[EPT], myRank[EPT];
  unsigned long long active = __ballot(1);
#pragma unroll
  for (int g = 0; g < EPT; ++g) {
    int i = base + wave * 512 + g * 32 + lane;  // groups processed IN ORDER
    uint32_t k = srcK[i];
    uint32_t p = srcI[i];
    uint32_t d = (k >> shift) & 0xFFu;
    unsigned long long mask = active;
#pragma unroll
    for (int b = 0; b < 8; ++b) {
      unsigned long long vote = __ballot((d >> b) & 1u);
      mask &= ((d >> b) & 1u) ? vote : ~vote;
    }
    int leaderLane = __ffsll((unsigned long long)mask) - 1;
    int rankInGrp  = __popcll(mask & ((1ull << lane) - 1ull));
    uint32_t cnt   = (uint32_t)__popcll(mask);
    uint32_t old = 0;
    if (lane == leaderLane) old = atomicAdd(&wdc[wave][d], cnt);
    old = (uint32_t)__shfl((int)old, leaderLane);
    myKey[g] = k; myPay[g] = p; myRank[g] = old + (uint32_t)rankInGrp;
  }
  __syncthreads();
  { // exclusive prefix over waves, one digit per thread
    uint32_t run = 0;
#pragma unroll
    for (int w = 0; w < 8; ++w) { exW[w][tid] = run; run += wdc[w][tid]; }
  }
  __syncthreads();
#pragma unroll
  for (int g = 0; g < EPT; ++g) {
    uint32_t d = (myKey[g] >> shift) & 0xFFu;
    uint32_t pos = histScanned[d * numTiles + blockIdx.x] + exW[wave][d] + myRank[g];
    dstK[pos] = myKey[g];
    dstI[pos] = myPay[g];
  }
}

// -------------------------- 3) float scan phase ----------------------------
__global__ __launch_bounds__(THREADS) void k_f_block_sums(
    const float* __restrict__ scores, const uint32_t* __restrict__ sidx,
    float* __restrict__ fBlock, int n) {
  __shared__ float rT[THREADS];
  int tid = threadIdx.x;
  int base = blockIdx.x * TILE;
  float s = 0.f;
#pragma unroll
  for (int k = 0; k < EPT; ++k) {
    int p = base + tid + k * THREADS;
    s += expf(scores[sidx[n - 1 - p]]);
  }
  rT[tid] = s;
  __syncthreads();
  for (int off = THREADS / 2; off > 0; off >>= 1) {
    if (tid < off) rT[tid] += rT[tid + off];
    __syncthreads();
  }
  if (tid == 0) fBlock[blockIdx.x] = rT[0];
}

__global__ __launch_bounds__(THREADS) void k_f_scan_tops(
    float* __restrict__ a, int len) {
  __shared__ float tsum[THREADS];
  int tid  = threadIdx.x;
  int base = tid * EPT;
  float v[EPT];
#pragma unroll
  for (int j = 0; j < EPT; ++j) v[j] = (base + j < len) ? a[base + j] : 0.f;
  float s = 0.f;
#pragma unroll
  for (int j = 0; j < EPT; ++j) { float x = v[j]; v[j] = s; s += x; }
  tsum[tid] = s;
  __syncthreads();
  for (int off = 1; off < THREADS; off <<= 1) {
    float t = (tid >= off) ? tsum[tid - off] : 0.f;
    __syncthreads();
    tsum[tid] += t;
    __syncthreads();
  }
  float texcl = tsum[tid] - s;
#pragma unroll
  for (int j = 0; j < EPT; ++j)
    if (base + j < len) a[base + j] = texcl + v[j];
}

// Per-block: async-to-LDS staging + WMMA-based 256-element tile scans + loss.
__global__ __launch_bounds__(THREADS) void k_scan_and_loss(
    const float* __restrict__ scores, const int* __restrict__ events,
    const uint32_t* __restrict__ sidx, const float* __restrict__ fTops,
    float* __restrict__ blockTerm, int* __restrict__ blockEvt, int n) {
  __shared__ uint32_t sidxT[TILE]; // sorted-index tile; later aliased as vals
  __shared__ float    lsc [TILE];  // raw score (async-gathered)
  __shared__ int      levI[TILE];  // event indicator (async-gathered)
  __shared__ float chunkTot[16];
  __shared__ float chunkOff[16];
  __shared__ float rT[THREADS];
  __shared__ int   rE[THREADS];
  float* vals = (float*)sidxT;     // exp(score); sidxT is dead by then

  int tid  = threadIdx.x;
  int wave = tid >> 5;
  int lane = tid & 31;
  int base = blockIdx.x * TILE;

  const uint32_t ldsSidx = lds_off(&sidxT[0]);
  const uint32_t ldsLsc  = lds_off(&lsc[0]);
  const uint32_t ldsLev  = lds_off(&levI[0]);

  // ---- stage 1: B128 async DMA of the contiguous (reversed-range) index tile
  const uint32_t* gsb = sidx + (n - TILE - base);
#pragma unroll
  for (int r = 0; r < 4; ++r) {
    int dw = r * 1024 + tid * 4;                    // this lane's 16B chunk
    unsigned long long ga = (unsigned long long)(uintptr_t)(gsb + dw);
    uint32_t la = ldsSidx + (uint32_t)dw * 4u;
    asm volatile("global_load_async_to_lds_b128 %0, %1, off"
                 :: "v"(la), "v"(ga) : "memory");
  }
  asm volatile("s_wait_asynccnt 0x0" ::: "memory");
  __syncthreads();

  // ---- stage 2: per-lane async *gathers* of scores/events straight into LDS
#pragma unroll
  for (int k = 0; k < EPT; ++k) {
    int lidx = tid + k * THREADS;
    uint32_t o = sidxT[TILE - 1 - lidx];            // reversed-sorted order
    unsigned long long gs = (unsigned long long)(uintptr_t)(scores + o);
    unsigned long long ge = (unsigned long long)(uintptr_t)(events + o);
    asm volatile("global_load_async_to_lds_b32 %0, %1, off"
                 :: "v"(ldsLsc + (uint32_t)lidx * 4u), "v"(gs) : "memory");
    asm volatile("global_load_async_to_lds_b32 %0, %1, off"
                 :: "v"(ldsLev + (uint32_t)lidx * 4u), "v"(ge) : "memory");
  }
  asm volatile("s_wait_asynccnt 0x0" ::: "memory");
  __syncthreads();                                  // sidxT consumed -> vals

  // ---- stage 3: exp(score)
#pragma unroll
  for (int k = 0; k < EPT; ++k) {
    int lidx = tid + k * THREADS;
    vals[lidx] = expf(lsc[lidx]);
  }
  __syncthreads();

  // ---- WMMA f32 16x16x4 tile scans: X(16x16) * U(upper-tri ones) ----------
  const int half = lane >> 4;       // 0: rows/K-even half, 1: other half
  const int nCol = lane & 15;       // doubles as A-row (M) and B-column (N)
#pragma unroll
  for (int cc = 0; cc < 2; ++cc) {
    int c  = wave + 8 * cc;         // this wave's chunk (256 elements)
    int cb = c * 256;
    v8f acc = {};
#pragma unroll
    for (int ki = 0; ki < 4; ++ki) {
      int k0 = 4 * ki + half * 2;
      v2f a, b;
      a.x = vals[cb + nCol * 16 + k0];       // A: X[M=nCol][k0], X[M][k0+1]
      a.y = vals[cb + nCol * 16 + k0 + 1];
      b.x = (k0     <= nCol) ? 1.f : 0.f;    // B: U[k0][N=nCol]
      b.y = (k0 + 1 <= nCol) ? 1.f : 0.f;
      acc = __builtin_amdgcn_wmma_f32_16x16x4_f32(
          false, a, false, b, (short)0, acc, false, false);
    }
    // row totals P[m][15] -> exclusive row offsets (16 shfls, all lanes)
    float offs[16];
    float run = 0.f;
#pragma unroll
    for (int j = 0; j < 16; ++j) {
      float tj = __shfl(acc[j & 7], ((j >> 3) << 4) + 15);
      offs[j] = run;
      run += tj;
    }
#pragma unroll
    for (int v = 0; v < 8; ++v) {
      int m = v + half * 8;
      float add = half ? offs[v + 8] : offs[v];
      vals[cb + m * 16 + nCol] = acc[v] + add;
    }
    if (lane == 0) chunkTot[c] = run;        // inclusive chunk total
  }
  __syncthreads();
  if (tid == 0) {
    float r = fTops[blockIdx.x];             // exclusive block base
#pragma unroll
    for (int c = 0; c < 16; ++c) { float t = chunkTot[c]; chunkOff[c] = r; r += t; }
  }
  __syncthreads();

  // ---- loss terms ----------------------------------------------------------
  float term = 0.f;
  int   ec   = 0;
#pragma unroll
  for (int k = 0; k < EPT; ++k) {
    int lidx = tid + k * THREADS;
    float Y  = vals[lidx] + chunkOff[lidx >> 8];   // inclusive prefix == suffix cumsum
    float m  = (levI[lidx] == 1) ? 1.f : 0.f;
    term += (-lsc[lidx] + logf(Y + EPSF)) * m;
    ec   += (levI[lidx] == 1) ? 1 : 0;
  }
  rT[tid] = term; rE[tid] = ec;
  __syncthreads();
  for (int off = THREADS / 2; off > 0; off >>= 1) {
    if (tid < off) { rT[tid] += rT[tid + off]; rE[tid] += rE[tid + off]; }
    __syncthreads();
  }
  if (tid == 0) { blockTerm[blockIdx.x] = rT[0]; blockEvt[blockIdx.x] = rE[0]; }
}

// ----------------------------- 4) final ------------------------------------
__global__ __launch_bounds__(THREADS) void k_final(
    const float* __restrict__ blockTerm, const int* __restrict__ blockEvt,
    float* __restrict__ out, int numTiles) {
  __shared__ float rT[THREADS];
  __shared__ int   rE[THREADS];
  int tid = threadIdx.x;
  float ts = 0.f; int ec = 0;
  for (int i = tid; i < numTiles; i += THREADS) { ts += blockTerm[i]; ec += blockEvt[i]; }
  rT[tid] = ts; rE[tid] = ec;
  __syncthreads();
  for (int off = THREADS / 2; off > 0; off >>= 1) {
    if (tid < off) { rT[tid] += rT[tid + off]; rE[tid] += rE[tid + off]; }
    __syncthreads();
  }
  if (tid == 0) {
    float ne = (float)rE[0];
    out[0] = (rE[0] > 0) ? rT[0] / fmaxf(ne, 1.f) : 0.f;
  }
}

// ---------------------------------------------------------------------------
extern "C" void kernel_launch(void* const* d_in, const int* in_sizes, int n_in,
                              void* d_out, int out_size, void* d_ws, size_t ws_size,
                              hipStream_t stream) {
  const float* scores = (const float*)d_in[0];
  const float* times  = (const float*)d_in[1];
  const int*   events = (const int*)d_in[2];
  int n = in_sizes[0];                       // 16,777,216 (divisible by TILE)

  int numTiles = n / TILE;                   // 4096
  int L        = RADIX * numTiles;           // 1,048,576 hist entries
  int topsLen  = L / TILE;                   // 256

  // workspace carve-out
  char* cur = (char*)d_ws;
  auto alloc = [&](size_t bytes) {
    void* p = (void*)cur;
    cur += (bytes + 255) & ~(size_t)255;
    return p;
  };
  uint32_t* keysA = (uint32_t*)alloc((size_t)n * 4);
  uint32_t* keysB = (uint32_t*)alloc((size_t)n * 4);
  uint32_t* idxA  = (uint32_t*)alloc((size_t)n * 4);
  uint32_t* idxB  = (uint32_t*)alloc((size_t)n * 4);
  uint32_t* hist  = (uint32_t*)alloc((size_t)L * 4);
  uint32_t* hTops = (uint32_t*)alloc((size_t)topsLen * 4);
  float*    fBlk  = (float*)   alloc((size_t)numTiles * 4);
  float*    bTerm = (float*)   alloc((size_t)numTiles * 4);
  int*      bEvt  = (int*)     alloc((size_t)numTiles * 4);

  k_init_keys<<<n / THREADS, THREADS, 0, stream>>>(times, keysA, idxA, n);

  for (int pass = 0; pass < 4; ++pass) {
    uint32_t* sK = (pass & 1) ? keysB : keysA;
    uint32_t* sI = (pass & 1) ? idxB  : idxA;
    uint32_t* dK = (pass & 1) ? keysA : keysB;
    uint32_t* dI = (pass & 1) ? idxA  : idxB;
    int shift = 8 * pass;
    k_radix_hist   <<<numTiles, THREADS, 0, stream>>>(sK, hist, numTiles, shift);
    k_scan_u32_tiles<<<L / TILE, THREADS, 0, stream>>>(hist, hTops);
    k_scan_u32_tops<<<1, 32, 0, stream>>>(hTops, topsLen);
    k_scan_u32_add <<<L / TILE, THREADS, 0, stream>>>(hist, hTops);
    k_radix_scatter<<<numTiles, THREADS, 0, stream>>>(sK, sI, dK, dI, hist,
                                                      numTiles, shift);
  }
  // after 4 passes sorted index lives in idxA

  k_f_block_sums<<<numTiles, THREADS, 0, stream>>>(scores, idxA, fBlk, n);
  k_f_scan_tops <<<1, THREADS, 0, stream>>>(fBlk, numTiles);
  k_scan_and_loss<<<numTiles, THREADS, 0, stream>>>(scores, events, idxA, fBlk,
                                                    bTerm, bEvt, n);
  k_final<<<1, THREADS, 0, stream>>>(bTerm, bEvt, (float*)d_out, numTiles);
}